// Decoder_52458730553797
// MI455X (gfx1250) — compile-verified
//
#include <hip/hip_runtime.h>
#include <hip/hip_bf16.h>
#include <math.h>

// Problem constants (match reference)
#define V_  32000
#define E_  512
#define H_  1024
#define S_  50
#define B_  64
#define T_  10
#define D2_ (2 * H_)   // 2048
#define SPLITK_ 4      // K-slices for the GRU-step GEMM (critical path)

typedef __attribute__((ext_vector_type(2))) float v2f;
typedef __attribute__((ext_vector_type(8))) float v8f;

// ---------------------------------------------------------------------------
// keys[b][s][d] = enc_out[s][b][d]   (S,B,2H) -> (B,S,2H)
// ---------------------------------------------------------------------------
__global__ void transpose_keys_kernel(const float* __restrict__ enc,
                                      float* __restrict__ keys) {
  size_t idx = (size_t)blockIdx.x * blockDim.x + threadIdx.x;
  size_t total = (size_t)S_ * B_ * D2_;
  if (idx >= total) return;
  int d = (int)(idx % D2_);
  size_t sb = idx / D2_;
  int b = (int)(sb % B_);
  int s = (int)(sb / B_);
  keys[((size_t)b * S_ + s) * D2_ + d] = enc[idx];
}

// ---------------------------------------------------------------------------
// xs[t*B+b][:] = emb[dec_in[b][t]], dec_in[b][0]=SOS=0, dec_in[b][t]=target[b][t-1]
// ---------------------------------------------------------------------------
__global__ void embed_kernel(const int* __restrict__ target,
                             const float* __restrict__ emb,
                             float* __restrict__ xs) {
  int m = blockIdx.x;              // t*B + b
  int t = m / B_;
  int b = m % B_;
  int tok = (t == 0) ? 0 : target[b * T_ + (t - 1)];
  const float* src = emb + (size_t)tok * E_;
  float* dst = xs + (size_t)m * E_;
  for (int i = threadIdx.x; i < E_; i += blockDim.x) dst[i] = src[i];
}

// ---------------------------------------------------------------------------
// WMMA f32 GEMM, 16x32 register-blocked:  C[M,N] = A[M,K] @ W[N,K]^T + bias[N]
// Each wave owns a 16x32 output tile (two adjacent 16x16 N-tiles sharing the
// A fragment) -> 4 independent v_wmma_f32_16x16x4_f32 accumulator chains and
// half the A-fragment traffic.
// Requires: M%16==0, N%32==0, K%8==0 (true for all call sites here).
//
// Fragment layouts per CDNA5 ISA 7.12.2:
//   A (16x4, MxK): lanes 0-15 hold K=k,k+1 in v[0],v[1]; lanes 16-31 K=k+2,k+3
//   B (4x16, KxN): column N = lane&15; half-waves split K exactly as A
//   C/D (16x16):   VGPR i -> row i (lanes 0-15) / row i+8 (lanes 16-31)
// ---------------------------------------------------------------------------
__global__ void wmma_gemm_bias_kernel(const float* __restrict__ A,
                                      const float* __restrict__ W,
                                      const float* __restrict__ bias,
                                      float* __restrict__ C,
                                      int M, int N, int K) {
  int wave = blockIdx.x * (blockDim.x >> 5) + (threadIdx.x >> 5);
  int lane = threadIdx.x & 31;
  int tilesN2 = N >> 5;                       // 16x32 tiles along N
  int tilesTotal = (M >> 4) * tilesN2;
  if (wave >= tilesTotal) return;             // wave-uniform: EXEC stays all-1s

  int row0 = (wave / tilesN2) << 4;
  int col0 = (wave % tilesN2) << 5;
  int lr = lane & 15;
  int hi = lane >> 4;

  const float* arow = A + (size_t)(row0 + lr) * K + 2 * hi;
  const float* w0   = W + (size_t)(col0 + lr) * K + 2 * hi;
  const float* w1   = W + (size_t)(col0 + 16 + lr) * K + 2 * hi;

  v8f accL0 = {}, accL1 = {}, accR0 = {}, accR1 = {};
  for (int k = 0; k < K; k += 8) {
    v2f a0  = *(const v2f*)(arow + k);
    v2f a1  = *(const v2f*)(arow + k + 4);
    v2f b00 = *(const v2f*)(w0 + k);
    v2f b01 = *(const v2f*)(w0 + k + 4);
    v2f b10 = *(const v2f*)(w1 + k);
    v2f b11 = *(const v2f*)(w1 + k + 4);
    accL0 = __builtin_amdgcn_wmma_f32_16x16x4_f32(false, a0, false, b00,
                                                  (short)0, accL0, false, false);
    accR0 = __builtin_amdgcn_wmma_f32_16x16x4_f32(false, a0, false, b10,
                                                  (short)0, accR0, false, false);
    accL1 = __builtin_amdgcn_wmma_f32_16x16x4_f32(false, a1, false, b01,
                                                  (short)0, accL1, false, false);
    accR1 = __builtin_amdgcn_wmma_f32_16x16x4_f32(false, a1, false, b11,
                                                  (short)0, accR1, false, false);
  }
  v8f accL = accL0 + accL1;
  v8f accR = accR0 + accR1;

  float bL = bias ? bias[col0 + lr] : 0.0f;
  float bR = bias ? bias[col0 + 16 + lr] : 0.0f;
#pragma unroll
  for (int i = 0; i < 8; ++i) {
    int r = row0 + i + 8 * hi;
    C[(size_t)r * N + col0 + lr]      = accL[i] + bL;
    C[(size_t)r * N + col0 + 16 + lr] = accR[i] + bR;
  }
}

// ---------------------------------------------------------------------------
// Split-K variant for the latency-critical GRU step GEMM (M=64 is too small
// to fill the chip otherwise). Wave (slice, tile) computes a 16x32 tile over
// K-range [slice*K/SPLITK, ...), writing partials to Cp[slice][M][N].
// Deterministic: partials are summed in fixed order by the gate kernel.
// ---------------------------------------------------------------------------
__global__ void wmma_gemm_splitk_kernel(const float* __restrict__ A,
                                        const float* __restrict__ W,
                                        float* __restrict__ Cp,  // [SPLITK][M][N]
                                        int M, int N, int K) {
  int wave = blockIdx.x * (blockDim.x >> 5) + (threadIdx.x >> 5);
  int lane = threadIdx.x & 31;
  int tilesN2 = N >> 5;
  int tilesTotal = (M >> 4) * tilesN2;
  if (wave >= tilesTotal * SPLITK_) return;   // wave-uniform

  int slice = wave / tilesTotal;
  int tile  = wave % tilesTotal;
  int row0 = (tile / tilesN2) << 4;
  int col0 = (tile % tilesN2) << 5;
  int lr = lane & 15;
  int hi = lane >> 4;
  int Kc = K / SPLITK_;
  int k0 = slice * Kc;

  const float* arow = A + (size_t)(row0 + lr) * K + k0 + 2 * hi;
  const float* w0   = W + (size_t)(col0 + lr) * K + k0 + 2 * hi;
  const float* w1   = W + (size_t)(col0 + 16 + lr) * K + k0 + 2 * hi;

  v8f accL0 = {}, accL1 = {}, accR0 = {}, accR1 = {};
  for (int k = 0; k < Kc; k += 8) {
    v2f a0  = *(const v2f*)(arow + k);
    v2f a1  = *(const v2f*)(arow + k + 4);
    v2f b00 = *(const v2f*)(w0 + k);
    v2f b01 = *(const v2f*)(w0 + k + 4);
    v2f b10 = *(const v2f*)(w1 + k);
    v2f b11 = *(const v2f*)(w1 + k + 4);
    accL0 = __builtin_amdgcn_wmma_f32_16x16x4_f32(false, a0, false, b00,
                                                  (short)0, accL0, false, false);
    accR0 = __builtin_amdgcn_wmma_f32_16x16x4_f32(false, a0, false, b10,
                                                  (short)0, accR0, false, false);
    accL1 = __builtin_amdgcn_wmma_f32_16x16x4_f32(false, a1, false, b01,
                                                  (short)0, accL1, false, false);
    accR1 = __builtin_amdgcn_wmma_f32_16x16x4_f32(false, a1, false, b11,
                                                  (short)0, accR1, false, false);
  }
  v8f accL = accL0 + accL1;
  v8f accR = accR0 + accR1;

  float* Cs = Cp + (size_t)slice * M * N;
#pragma unroll
  for (int i = 0; i < 8; ++i) {
    int r = row0 + i + 8 * hi;
    Cs[(size_t)r * N + col0 + lr]      = accL[i];
    Cs[(size_t)r * N + col0 + 16 + lr] = accR[i];
  }
}

// ---------------------------------------------------------------------------
// Fused GRU gate update for one time step; sums the SPLITK gh partials + b_hh.
// gi: [B,3H], gh_part: [SPLITK][B,3H]. hprev, hout: [B,H].
// ---------------------------------------------------------------------------
__global__ void gru_gate_kernel(const float* __restrict__ gi,
                                const float* __restrict__ gh_part,
                                const float* __restrict__ b_hh,
                                const float* __restrict__ hprev,
                                float* __restrict__ hout) {
  int idx = blockIdx.x * blockDim.x + threadIdx.x;
  if (idx >= B_ * H_) return;
  int b = idx / H_;
  int j = idx % H_;
  size_t base = (size_t)b * 3 * H_;
  const float* gib = gi + base;

  float hr = b_hh[j], hz = b_hh[H_ + j], hn = b_hh[2 * H_ + j];
#pragma unroll
  for (int s = 0; s < SPLITK_; ++s) {
    const float* ghs = gh_part + (size_t)s * B_ * 3 * H_ + base;
    hr += ghs[j];
    hz += ghs[H_ + j];
    hn += ghs[2 * H_ + j];
  }

  float ir = gib[j], iz = gib[H_ + j], in = gib[2 * H_ + j];
  float r = 1.0f / (1.0f + __expf(-(ir + hr)));
  float z = 1.0f / (1.0f + __expf(-(iz + hz)));
  float n = tanhf(in + r * hn);
  float h = hprev[idx];
  hout[idx] = (1.0f - z) * n + z * h;
}

// ---------------------------------------------------------------------------
// scores[m][s] = sum_h tanh(wq[m][h] + uk[b][s][h]) * va_w[h] + va_b,  m = t*B+b
// ---------------------------------------------------------------------------
__global__ void attn_scores_kernel(const float* __restrict__ wq,
                                   const float* __restrict__ uk,
                                   const float* __restrict__ va_w,
                                   const float* __restrict__ va_b,
                                   float* __restrict__ scores) {
  __shared__ float s_wq[H_];
  __shared__ float s_va[H_];
  __shared__ float red[256];
  int m = blockIdx.x;
  int b = m % B_;
  for (int i = threadIdx.x; i < H_; i += blockDim.x) {
    s_wq[i] = wq[(size_t)m * H_ + i];
    s_va[i] = va_w[i];
  }
  __syncthreads();
  float vb = va_b[0];
  for (int s = 0; s < S_; ++s) {
    const float* ukr = uk + ((size_t)b * S_ + s) * H_;
    float acc = 0.0f;
    for (int i = threadIdx.x; i < H_; i += blockDim.x)
      acc += tanhf(s_wq[i] + ukr[i]) * s_va[i];
    red[threadIdx.x] = acc;
    __syncthreads();
    for (int off = blockDim.x >> 1; off > 0; off >>= 1) {
      if (threadIdx.x < (unsigned)off) red[threadIdx.x] += red[threadIdx.x + off];
      __syncthreads();
    }
    if (threadIdx.x == 0) scores[(size_t)m * S_ + s] = red[0] + vb;
    __syncthreads();
  }
}

// ---------------------------------------------------------------------------
// In-place softmax over S per row; one wave32 per row.
// ---------------------------------------------------------------------------
__global__ void softmax_rows_kernel(float* __restrict__ scores) {
  int m = blockIdx.x;
  float* row = scores + (size_t)m * S_;
  int lane = threadIdx.x;
  float mx = -INFINITY;
  for (int s = lane; s < S_; s += 32) mx = fmaxf(mx, row[s]);
  for (int off = 16; off > 0; off >>= 1) mx = fmaxf(mx, __shfl_xor(mx, off));
  float sum = 0.0f;
  for (int s = lane; s < S_; s += 32) {
    float e = __expf(row[s] - mx);
    row[s] = e;
    sum += e;
  }
  for (int off = 16; off > 0; off >>= 1) sum += __shfl_xor(sum, off);
  float inv = 1.0f / sum;
  for (int s = lane; s < S_; s += 32) row[s] *= inv;
}

// ---------------------------------------------------------------------------
// context[b][t][d] = sum_s weights[t*B+b][s] * keys[b][s][d]; into d_out.
// ---------------------------------------------------------------------------
__global__ void context_kernel(const float* __restrict__ weights,
                               const float* __restrict__ keys,
                               float* __restrict__ out) {
  __shared__ float s_w[64];
  int bt = blockIdx.x;
  int b = bt / T_;
  int t = bt % T_;
  const float* wrow = weights + ((size_t)t * B_ + b) * S_;
  for (int i = threadIdx.x; i < S_; i += blockDim.x) s_w[i] = wrow[i];
  __syncthreads();
  const float* kb = keys + (size_t)b * S_ * D2_;
  float* orow = out + ((size_t)b * T_ + t) * D2_;
  for (int d = threadIdx.x; d < D2_; d += blockDim.x) {
    float acc = 0.0f;
    for (int s = 0; s < S_; ++s) acc += s_w[s] * kb[(size_t)s * D2_ + d];
    orow[d] = acc;
  }
}

// ---------------------------------------------------------------------------
// Stable in-place log-softmax over D2 per row.
// ---------------------------------------------------------------------------
__global__ void log_softmax_rows_kernel(float* __restrict__ out) {
  __shared__ float red[256];
  int m = blockIdx.x;
  float* row = out + (size_t)m * D2_;
  float mx = -INFINITY;
  for (int i = threadIdx.x; i < D2_; i += blockDim.x) mx = fmaxf(mx, row[i]);
  red[threadIdx.x] = mx;
  __syncthreads();
  for (int off = 128; off > 0; off >>= 1) {
    if (threadIdx.x < (unsigned)off)
      red[threadIdx.x] = fmaxf(red[threadIdx.x], red[threadIdx.x + off]);
    __syncthreads();
  }
  mx = red[0];
  __syncthreads();
  float sum = 0.0f;
  for (int i = threadIdx.x; i < D2_; i += blockDim.x) sum += __expf(row[i] - mx);
  red[threadIdx.x] = sum;
  __syncthreads();
  for (int off = 128; off > 0; off >>= 1) {
    if (threadIdx.x < (unsigned)off) red[threadIdx.x] += red[threadIdx.x + off];
    __syncthreads();
  }
  float lse = mx + __logf(red[0]);
  __syncthreads();
  for (int i = threadIdx.x; i < D2_; i += blockDim.x) row[i] -= lse;
}

__global__ void copy_kernel(const float* __restrict__ src,
                            float* __restrict__ dst, int n) {
  int i = blockIdx.x * blockDim.x + threadIdx.x;
  if (i < n) dst[i] = src[i];
}

// ---------------------------------------------------------------------------
// Host launcher
// ---------------------------------------------------------------------------
extern "C" void kernel_launch(void* const* d_in, const int* in_sizes, int n_in,
                              void* d_out, int out_size, void* d_ws, size_t ws_size,
                              hipStream_t stream) {
  (void)in_sizes; (void)n_in; (void)out_size; (void)ws_size;

  const float* enc_out = (const float*)d_in[0];   // [S,B,2H]
  const float* hidden  = (const float*)d_in[1];   // [1,B,H]
  const int*   target  = (const int*)  d_in[2];   // [B,T]
  const float* emb     = (const float*)d_in[3];   // [V,E]
  const float* w_ih    = (const float*)d_in[4];   // [3H,E]
  const float* w_hh    = (const float*)d_in[5];   // [3H,H]
  const float* b_ih    = (const float*)d_in[6];   // [3H]
  const float* b_hh    = (const float*)d_in[7];   // [3H]
  const float* wa_w    = (const float*)d_in[8];   // [H,H]
  const float* wa_b    = (const float*)d_in[9];   // [H]
  const float* ua_w    = (const float*)d_in[10];  // [H,2H]
  const float* ua_b    = (const float*)d_in[11];  // [H]
  const float* va_w    = (const float*)d_in[12];  // [1,H]
  const float* va_b    = (const float*)d_in[13];  // [1]

  float* out = (float*)d_out;                     // [B,T,2H] ++ [B,H]

  // Workspace carve-up (floats)
  float* ws      = (float*)d_ws;
  float* keys    = ws;                                   // B*S*2H        = 6,553,600
  float* uk      = keys    + (size_t)B_ * S_ * D2_;      // B*S*H         = 3,276,800
  float* xs      = uk      + (size_t)B_ * S_ * H_;       // T*B*E         =   327,680
  float* gi_all  = xs      + (size_t)T_ * B_ * E_;       // T*B*3H        = 1,966,080
  float* gh_part = gi_all  + (size_t)T_ * B_ * 3 * H_;   // SPLITK*B*3H   =   786,432
  float* hs      = gh_part + (size_t)SPLITK_ * B_ * 3 * H_; // T*B*H      =   655,360
  float* wq      = hs      + (size_t)T_ * B_ * H_;       // T*B*H         =   655,360
  float* scores  = wq      + (size_t)T_ * B_ * H_;       // T*B*S         =    32,000

  // 1) enc_out -> keys
  {
    size_t total = (size_t)S_ * B_ * D2_;
    transpose_keys_kernel<<<dim3((unsigned)((total + 255) / 256)), dim3(256), 0, stream>>>(
        enc_out, keys);
  }

  // 2) embedding gather -> xs [T*B, E]
  embed_kernel<<<dim3(T_ * B_), dim3(128), 0, stream>>>(target, emb, xs);

  // 3) gi_all = xs @ w_ih^T + b_ih   (M=640, N=3072, K=512)
  {
    int M = T_ * B_, N = 3 * H_, K = E_;
    int tiles = (M / 16) * (N / 32);
    wmma_gemm_bias_kernel<<<dim3((tiles + 7) / 8), dim3(256), 0, stream>>>(
        xs, w_ih, b_ih, gi_all, M, N, K);
  }

  // 4) uk = keys @ ua_w^T + ua_b     (M=3200, N=1024, K=2048)
  {
    int M = B_ * S_, N = H_, K = D2_;
    int tiles = (M / 16) * (N / 32);
    wmma_gemm_bias_kernel<<<dim3((tiles + 7) / 8), dim3(256), 0, stream>>>(
        keys, ua_w, ua_b, uk, M, N, K);
  }

  // 5) GRU scan: 10 sequential {split-K gh GEMM, fused gate+reduce} pairs
  for (int t = 0; t < T_; ++t) {
    const float* hprev = (t == 0) ? hidden : (hs + (size_t)(t - 1) * B_ * H_);
    float* hout = hs + (size_t)t * B_ * H_;
    {
      int M = B_, N = 3 * H_, K = H_;
      int waves = (M / 16) * (N / 32) * SPLITK_;
      wmma_gemm_splitk_kernel<<<dim3((waves + 7) / 8), dim3(256), 0, stream>>>(
          hprev, w_hh, gh_part, M, N, K);
    }
    gru_gate_kernel<<<dim3((B_ * H_ + 255) / 256), dim3(256), 0, stream>>>(
        gi_all + (size_t)t * B_ * 3 * H_, gh_part, b_hh, hprev, hout);
  }

  // 6) wq = hs @ wa_w^T + wa_b       (M=640, N=1024, K=1024)
  {
    int M = T_ * B_, N = H_, K = H_;
    int tiles = (M / 16) * (N / 32);
    wmma_gemm_bias_kernel<<<dim3((tiles + 7) / 8), dim3(256), 0, stream>>>(
        hs, wa_w, wa_b, wq, M, N, K);
  }

  // 7) additive attention scores
  attn_scores_kernel<<<dim3(T_ * B_), dim3(256), 0, stream>>>(
      wq, uk, va_w, va_b, scores);

  // 8) softmax over S (in place -> weights)
  softmax_rows_kernel<<<dim3(T_ * B_), dim3(32), 0, stream>>>(scores);

  // 9) context = einsum('tbs,bsd->btd') written directly into d_out
  context_kernel<<<dim3(B_ * T_), dim3(256), 0, stream>>>(scores, keys, out);

  // 10) log_softmax over 2H, in place on d_out
  log_softmax_rows_kernel<<<dim3(B_ * T_), dim3(256), 0, stream>>>(out);

  // 11) h_last = hs[T-1] appended after out
  copy_kernel<<<dim3((B_ * H_ + 255) / 256), dim3(256), 0, stream>>>(
      hs + (size_t)(T_ - 1) * B_ * H_, out + (size_t)B_ * T_ * D2_, B_ * H_);
}